// FeatureTokenizer_19722489823864
// MI455X (gfx1250) — compile-verified
//
#include <hip/hip_runtime.h>

typedef __attribute__((ext_vector_type(2))) float v2f;
typedef __attribute__((ext_vector_type(4))) float v4f;
typedef __attribute__((ext_vector_type(8))) float v8f;

#define FT_BATCH   16384
#define FT_NFEAT   128
#define FT_DTOK    192
#define FT_NTOK    129              // 1 cls + 128 feature tokens
#define FT_ROWSTR  (FT_NTOK * FT_DTOK)  // 24768 floats per batch row

// tokens[b, 1+f, d] = x[b,f] * W[f,d] + bias[f,d], computed as
// D = A(16x4: M=d, K=0 holds W-row) x B(4x16: K=0 row holds x-column) + C(bias bcast)
// D layout (M=d orientation): lane n<16 -> batch col n, VGPR i -> d = d0+i;
// lanes 16-31 -> batch col n-16, VGPR i -> d = d0+8+i. So each lane holds 8
// CONSECUTIVE d values of one batch row -> two aligned b128 NT stores.
__global__ __launch_bounds__(256) void FeatureTokenizer_wmma_kernel(
    const float* __restrict__ x,     // [BATCH, NFEAT]
    const float* __restrict__ W,     // [NFEAT, DTOK]
    const float* __restrict__ bias,  // [NFEAT, DTOK]
    float* __restrict__ out)         // [BATCH, NTOK, DTOK]
{
    const int lane = threadIdx.x & 31;
    const int wave = threadIdx.x >> 5;
    const int f    = blockIdx.y * 8 + wave;   // one feature per wave
    const int b0   = blockIdx.x * 16;         // 16-batch tile
    const int l15  = lane & 15;
    const bool lo  = lane < 16;
    const int mb   = lo ? 0 : 8;              // d sub-offset for this half-wave

    // B operand (4x16): only the K=0 row carries data = x column for feature f.
    // VGPR0: lanes0-15 = K0 row (N=batch), lanes16-31 = K1 row (zero).
    const float xv = x[(size_t)(b0 + l15) * FT_NFEAT + f];
    v2f bmat;
    bmat.x = lo ? xv : 0.0f;
    bmat.y = 0.0f;

    const float* wrow = W    + (size_t)f * FT_DTOK;
    const float* brow = bias + (size_t)f * FT_DTOK;
    float*       obase = out + (size_t)(1 + f) * FT_DTOK
                             + (size_t)(b0 + l15) * FT_ROWSTR + mb;

    for (int d0 = 0; d0 < FT_DTOK; d0 += 16) {
        // A operand (16x4): lanes0-15 VGPR0 = K0 (W row slice), VGPR1 = K1 (0);
        // lanes16-31 VGPR0/1 = K2/K3 (0).
        const float wv = wrow[d0 + l15];
        v2f amat;
        amat.x = lo ? wv : 0.0f;
        amat.y = 0.0f;

        // C operand: bias broadcast across batch; per lane = 8 consecutive d.
        const v4f* bp = (const v4f*)(brow + d0 + mb);
        v8f c;
        c.lo = bp[0];
        c.hi = bp[1];

        v8f d = __builtin_amdgcn_wmma_f32_16x16x4_f32(
            /*neg_a=*/false, amat, /*neg_b=*/false, bmat,
            /*c_mod=*/(short)0, c, /*reuse_a=*/false, /*reuse_b=*/false);

        float* dst = obase + d0;
        __builtin_nontemporal_store(d.lo, (v4f*)dst);        // 16B
        __builtin_nontemporal_store(d.hi, (v4f*)(dst + 4));  // 16B
    }
}

// out[b, 0, :] = cls_token  — pure streaming broadcast, NT b128 stores.
// One float4 per thread; BATCH * (192/4) = 786432 float4s exactly.
__global__ __launch_bounds__(256) void FeatureTokenizer_cls_kernel(
    const float* __restrict__ cls, float* __restrict__ out)
{
    const int idx = blockIdx.x * 256 + threadIdx.x;
    const int row = idx / 48;          // batch row
    const int j   = idx - row * 48;    // float4 index within d_token
    v4f v = ((const v4f*)cls)[j];
    __builtin_nontemporal_store(v, (v4f*)(out + (size_t)row * FT_ROWSTR) + j);
}

extern "C" void kernel_launch(void* const* d_in, const int* in_sizes, int n_in,
                              void* d_out, int out_size, void* d_ws, size_t ws_size,
                              hipStream_t stream) {
    const float* x    = (const float*)d_in[0];  // [16384,128]
    const float* W    = (const float*)d_in[1];  // [128,192]
    const float* bias = (const float*)d_in[2];  // [128,192]
    const float* cls  = (const float*)d_in[3];  // [192]
    float* out = (float*)d_out;                 // [16384,129,192]

    dim3 grid(FT_BATCH / 16, FT_NFEAT / 8, 1);
    FeatureTokenizer_wmma_kernel<<<grid, 256, 0, stream>>>(x, W, bias, out);

    const int cls_threads = FT_BATCH * (FT_DTOK / 4);       // 786432
    FeatureTokenizer_cls_kernel<<<cls_threads / 256, 256, 0, stream>>>(cls, out);
}